// VectorQuantizer_63316407877789
// MI455X (gfx1250) — compile-verified
//
#include <hip/hip_runtime.h>
#include <cfloat>
#include <cmath>

typedef __attribute__((ext_vector_type(2))) float v2f;
typedef __attribute__((ext_vector_type(8))) float v8f;

#define KCODES 1024
#define DDIM   256
#define NVEC   32768          // 32*32*32 vectors
#define QELEMS 8388608        // 32*256*32*32
#define LDS_PITCH 260         // 256 + 4 pad: conflict-free b64 lane reads, 16B rows

#if __has_builtin(__builtin_amdgcn_global_load_async_to_lds_b128)
#define USE_ASYNC_LDS 1
typedef int v4i_ __attribute__((vector_size(16)));             // matches builtin param pointee
typedef __attribute__((address_space(1))) v4i_* as1_v4i;       // global int4*
typedef __attribute__((address_space(3))) v4i_* as3_v4i;       // LDS int4*
#else
#define USE_ASYNC_LDS 0
#endif

__device__ __forceinline__ void wait_async_then_barrier() {
#if USE_ASYNC_LDS
#if __has_builtin(__builtin_amdgcn_s_wait_asynccnt)
    __builtin_amdgcn_s_wait_asynccnt(0);
#else
    asm volatile("s_wait_asynccnt 0" ::: "memory");
#endif
#endif
    __syncthreads();
}

// ---------------- K0: zero histogram + loss accumulator ----------------
__global__ void k0_init(float* __restrict__ counts, float* __restrict__ acc) {
    int t = threadIdx.x;
    #pragma unroll
    for (int i = 0; i < 4; ++i) counts[t + i * 256] = 0.0f;
    if (t == 0) acc[0] = 0.0f;
}

// ---------------- K1: NCHW -> [N][D] row-major pack (tiled transpose) ----
__global__ void k1_pack(const float* __restrict__ z, float* __restrict__ zp) {
    __shared__ float tile[32][33];
    int blk = blockIdx.x;            // b*256 + dt*32 + st
    int st = blk & 31; blk >>= 5;
    int dt = blk & 7;  blk >>= 3;
    int b  = blk;
    int t  = threadIdx.x;
    int tr = t >> 5, tc = t & 31;
    size_t zb = ((size_t)b * 256 + (size_t)dt * 32) * 1024 + (size_t)st * 32;
    #pragma unroll
    for (int i = 0; i < 4; ++i) {
        int dl = i * 8 + tr;
        tile[dl][tc] = z[zb + (size_t)dl * 1024 + tc];   // coalesced along hw
    }
    __syncthreads();
    size_t ob = ((size_t)b * 1024 + (size_t)st * 32) * 256 + (size_t)dt * 32;
    #pragma unroll
    for (int i = 0; i < 4; ++i) {
        int hl = i * 8 + tr;
        zp[ob + (size_t)hl * 256 + tc] = tile[tc][hl];   // coalesced along d
    }
}

// ---------------- K2: codebook squared norms ----------------
__global__ void k2_cnorm(const float* __restrict__ cb, float* __restrict__ cn) {
    int row  = blockIdx.x * 8 + (threadIdx.x >> 5);
    int lane = threadIdx.x & 31;
    const float* r = cb + (size_t)row * DDIM;
    float s = 0.f;
    #pragma unroll
    for (int i = 0; i < 8; ++i) { float v = r[lane + i * 32]; s += v * v; }
    #pragma unroll
    for (int off = 16; off; off >>= 1) s += __shfl_xor(s, off, 32);
    if (lane == 0) cn[row] = s;
}

// ---------------- K3: WMMA distance GEMM + fused argmin ----------------
// 8 waves/block; each wave owns a 16-row z block (A in VGPRs), all waves share
// double-buffered codebook tiles in LDS (B), staged via async global->LDS DMA.
// score = ||c||^2 - 2 z.c  (z-norm constant per row, irrelevant for argmin)
__global__ void __launch_bounds__(256) k3_argmin(
        const float* __restrict__ zp, const float* __restrict__ cb,
        const float* __restrict__ cn, int* __restrict__ outIdx) {
    __shared__ float bufB[2][16 * LDS_PITCH];
    const int t     = threadIdx.x;
    const int lane  = t & 31;
    const int wave  = t >> 5;
    const int lmod  = lane & 15;
    const int lhalf = lane >> 4;
    const int row0  = (blockIdx.x * 8 + wave) * 16;

    // A fragments: 16x4 f32 layout => lanes 0-15 hold (K=0,1), 16-31 hold (K=2,3)
    const float* zrow = zp + (size_t)(row0 + lmod) * DDIM + lhalf * 2;
    v2f afrag[64];
    #pragma unroll
    for (int k = 0; k < 64; ++k)
        afrag[k] = *reinterpret_cast<const v2f*>(zrow + 4 * k);

    float bv[8]; int bi[8];
    #pragma unroll
    for (int s = 0; s < 8; ++s) { bv[s] = FLT_MAX; bi[s] = 0; }

#if !USE_ASYNC_LDS
    float4 stage[4];
#endif

    // ---- stage tile 0 into buffer 0 ----
#if USE_ASYNC_LDS
    #pragma unroll
    for (int i = 0; i < 4; ++i) {
        int f4 = i * 256 + t;                          // 1024 float4 per 16KB tile
        const float* g = cb + (size_t)(f4 >> 6) * DDIM + (f4 & 63) * 4;
        float* l = &bufB[0][(f4 >> 6) * LDS_PITCH + (f4 & 63) * 4];
        __builtin_amdgcn_global_load_async_to_lds_b128((as1_v4i)g, (as3_v4i)l, 0, 0);
    }
    wait_async_then_barrier();
#else
    #pragma unroll
    for (int i = 0; i < 4; ++i) {
        int f4 = i * 256 + t;
        stage[i] = *reinterpret_cast<const float4*>(
            cb + (size_t)(f4 >> 6) * DDIM + (f4 & 63) * 4);
    }
    #pragma unroll
    for (int i = 0; i < 4; ++i) {
        int f4 = i * 256 + t;
        *reinterpret_cast<float4*>(&bufB[0][(f4 >> 6) * LDS_PITCH + (f4 & 63) * 4]) = stage[i];
    }
    __syncthreads();
#endif

    const int boff = lmod * LDS_PITCH + lhalf * 2;

    for (int tt = 0; tt < 64; ++tt) {
        // ---- kick off DMA of next tile into the other buffer ----
        if (tt + 1 < 64) {
            const float* src = cb + (size_t)(tt + 1) * 16 * DDIM;
            float* dst = bufB[(tt + 1) & 1];
#if USE_ASYNC_LDS
            #pragma unroll
            for (int i = 0; i < 4; ++i) {
                int f4 = i * 256 + t;
                const float* g = src + (size_t)(f4 >> 6) * DDIM + (f4 & 63) * 4;
                float* l = dst + (f4 >> 6) * LDS_PITCH + (f4 & 63) * 4;
                __builtin_amdgcn_global_load_async_to_lds_b128((as1_v4i)g, (as3_v4i)l, 0, 0);
            }
#else
            #pragma unroll
            for (int i = 0; i < 4; ++i) {
                int f4 = i * 256 + t;
                stage[i] = *reinterpret_cast<const float4*>(
                    src + (size_t)(f4 >> 6) * DDIM + (f4 & 63) * 4);
            }
#endif
        }

        // ---- 64 WMMAs over this tile, two independent accumulator chains ----
        const float* bp = bufB[tt & 1] + boff;
        v8f acc0 = {0.f, 0.f, 0.f, 0.f, 0.f, 0.f, 0.f, 0.f};
        v8f acc1 = {0.f, 0.f, 0.f, 0.f, 0.f, 0.f, 0.f, 0.f};
        #pragma unroll
        for (int k = 0; k < 64; k += 2) {
            v2f b0 = *reinterpret_cast<const v2f*>(bp + 4 * k);
            v2f b1 = *reinterpret_cast<const v2f*>(bp + 4 * k + 4);
            acc0 = __builtin_amdgcn_wmma_f32_16x16x4_f32(
                false, afrag[k],     false, b0, (short)0, acc0, false, false);
            acc1 = __builtin_amdgcn_wmma_f32_16x16x4_f32(
                false, afrag[k + 1], false, b1, (short)0, acc1, false, false);
        }

        float c2 = cn[tt * 16 + lmod];
        int idxb = tt * 16 + lmod;
        #pragma unroll
        for (int s = 0; s < 8; ++s) {
            float sc = c2 - 2.0f * (acc0[s] + acc1[s]);
            if (sc < bv[s]) { bv[s] = sc; bi[s] = idxb; }
        }

#if USE_ASYNC_LDS
        wait_async_then_barrier();
#else
        __syncthreads();           // reads of buf[tt&1] done everywhere
        if (tt + 1 < 64) {
            float* dst = bufB[(tt + 1) & 1];
            #pragma unroll
            for (int i = 0; i < 4; ++i) {
                int f4 = i * 256 + t;
                *reinterpret_cast<float4*>(&dst[(f4 >> 6) * LDS_PITCH + (f4 & 63) * 4]) = stage[i];
            }
            __syncthreads();
        }
#endif
    }

    // min+index reduce within each 16-lane half (rows s / s+8)
    #pragma unroll
    for (int s = 0; s < 8; ++s) {
        #pragma unroll
        for (int off = 8; off; off >>= 1) {
            float ov = __shfl_xor(bv[s], off, 32);
            int   oi = __shfl_xor(bi[s], off, 32);
            if (ov < bv[s] || (ov == bv[s] && oi < bi[s])) { bv[s] = ov; bi[s] = oi; }
        }
    }
    if (lmod == 0) {
        #pragma unroll
        for (int s = 0; s < 8; ++s)
            outIdx[row0 + lhalf * 8 + s] = bi[s];
    }
}

// ---------------- K4: gather quantized (NCHW) + loss partials ----------------
__global__ void k4_quant_loss(const float* __restrict__ z, const float* __restrict__ cb,
                              const int* __restrict__ idx, float* __restrict__ qout,
                              float* __restrict__ acc) {
    __shared__ float qtile[32 * LDS_PITCH];
    __shared__ int   widx[32];
    int bh = blockIdx.x;             // b*32 + h
    int t  = threadIdx.x;
    if (t < 32) widx[t] = idx[bh * 32 + t];
    __syncthreads();
    #pragma unroll
    for (int i = 0; i < 8; ++i) {    // stage 32 codebook rows (coalesced)
        int f4 = i * 256 + t;
        int row = f4 >> 6, c4 = f4 & 63;
        float4 v = *reinterpret_cast<const float4*>(cb + (size_t)widx[row] * DDIM + c4 * 4);
        *reinterpret_cast<float4*>(&qtile[row * LDS_PITCH + c4 * 4]) = v;
    }
    __syncthreads();
    int w = t & 31, wv = t >> 5;
    size_t zb = (size_t)(bh >> 5) * 262144 + (size_t)(bh & 31) * 32 + w;
    float lsum = 0.f;
    #pragma unroll 4
    for (int i = 0; i < 32; ++i) {
        int d = i * 8 + wv;
        float q = qtile[w * LDS_PITCH + d];
        size_t o = zb + (size_t)d * 1024;
        float zz = z[o];
        float df = q - zz;
        lsum += df * df;
        qout[o] = q;                 // coalesced along w
    }
    #pragma unroll
    for (int off = 16; off; off >>= 1) lsum += __shfl_xor(lsum, off, 32);
    if (w == 0) atomicAdd(acc, lsum);
}

// ---------------- K5: one-hot encodings + histogram ----------------
__global__ void k5_enc(const int* __restrict__ idx, float* __restrict__ enc,
                       float* __restrict__ counts) {
    int t = threadIdx.x;
    int lane = t & 31;
    int n = blockIdx.x * 8 + (t >> 5);
    int id = idx[n];
    float* row = enc + (size_t)n * 1024;
    #pragma unroll
    for (int i = 0; i < 16; ++i) {
        int j0 = i * 64 + lane * 2;
        float2 v;
        v.x = (j0     == id) ? 1.0f : 0.0f;
        v.y = (j0 + 1 == id) ? 1.0f : 0.0f;
        *reinterpret_cast<float2*>(row + j0) = v;   // 8B-aligned region
    }
    if (lane == 0) atomicAdd(&counts[id], 1.0f);
}

// ---------------- K6: loss + perplexity scalars ----------------
__global__ void k6_final(const float* __restrict__ counts, const float* __restrict__ acc,
                         float* __restrict__ scal) {
    int lane = threadIdx.x & 31;
    float s = 0.f;
    #pragma unroll
    for (int i = 0; i < 32; ++i) {
        float p = counts[lane + i * 32] * (1.0f / 32768.0f);
        s += p * logf(p + 1e-10f);
    }
    #pragma unroll
    for (int off = 16; off; off >>= 1) s += __shfl_xor(s, off, 32);
    if (lane == 0) {
        scal[0] = acc[0] * (1.25f / 8388608.0f);  // q_loss + beta*e_loss = 1.25*MSE
        scal[1] = expf(-s);
    }
}

extern "C" void kernel_launch(void* const* d_in, const int* in_sizes, int n_in,
                              void* d_out, int out_size, void* d_ws, size_t ws_size,
                              hipStream_t stream) {
    const float* z  = (const float*)d_in[0];   // [32,256,32,32]
    const float* cb = (const float*)d_in[1];   // [1024,256]
    float* out   = (float*)d_out;
    float* qout  = out;                        // 8388608 floats
    float* scal  = out + QELEMS;               // loss, perplexity
    float* enc   = out + QELEMS + 2;           // [32768,1024]
    // packed-z scratch lives inside the encodings region (16B aligned),
    // consumed by k3 before k5 overwrites the whole region.
    float* zpack = out + QELEMS + 4;

    float* cnorm   = (float*)d_ws;             // 1024 f
    float* counts  = cnorm + 1024;             // 1024 f
    float* acc     = counts + 1024;            // 1 f
    int*   indices = (int*)((char*)d_ws + 16384);  // 32768 ints

    k0_init      <<<1,    256, 0, stream>>>(counts, acc);
    k1_pack      <<<8192, 256, 0, stream>>>(z, zpack);
    k2_cnorm     <<<128,  256, 0, stream>>>(cb, cnorm);
    k3_argmin    <<<256,  256, 0, stream>>>(zpack, cb, cnorm, indices);
    k4_quant_loss<<<1024, 256, 0, stream>>>(z, cb, indices, qout, acc);
    k5_enc       <<<4096, 256, 0, stream>>>(indices, enc, counts);
    k6_final     <<<1,    32,  0, stream>>>(counts, acc, scal);
}